// Set2SetNet_76776835383347
// MI455X (gfx1250) — compile-verified
//
#include <hip/hip_runtime.h>
#include <hip/hip_bf16.h>

// ---------------------------------------------------------------------------
// Set2SetNet for MI455X (gfx1250, wave32).
//   L1: agg1[dst] += x[src]; deg[dst]++           (f32 global atomics, L2-resident)
//       h1 = relu([agg1/deg | x] @ [W1l;W1r]^T + b1l)   via V_WMMA_F32_16X16X4_F32
//   L2: agg2[dst] += h1[src]
//       h2 = relu([agg2/deg | h1] @ [W2l;W2r]^T + b2l)  via V_WMMA_F32_16X16X4_F32
//   B matrices are prepacked in WMMA lane order (zero-padded) so the WMMA inner
//   loop has zero divergent loads.
//   Set2Set: one block per graph (batch sorted -> contiguous segments),
//            4 LSTM+attention steps in shared memory, fused final FC.
// ---------------------------------------------------------------------------

typedef __attribute__((ext_vector_type(2))) float v2f;
typedef __attribute__((ext_vector_type(8))) float v8f;

#define N_NODES  200000
#define N_EDGES  6400000
#define N_GRAPHS 1024
#define F_IN     16
#define H1       8
#define H2       16
#define STEPS    4
#define N_TILES  (N_NODES / 16)   // 12500, exact

// ---------------- init kernels ----------------------------------------------
__global__ void zero_f(float* __restrict__ p, int n) {
    int i = blockIdx.x * blockDim.x + threadIdx.x;
    if (i < n) p[i] = 0.0f;
}

__global__ void init_seg(int* __restrict__ gstart, int* __restrict__ gend) {
    int g = blockIdx.x * blockDim.x + threadIdx.x;
    if (g < N_GRAPHS) { gstart[g] = N_NODES; gend[g] = 0; }
}

__global__ void seg_bounds(const int* __restrict__ batch,
                           int* __restrict__ gstart, int* __restrict__ gend) {
    int i = blockIdx.x * blockDim.x + threadIdx.x;
    if (i < N_NODES) {
        int b = batch[i];
        atomicMin(&gstart[b], i);
        atomicMax(&gend[b], i + 1);
    }
}

// ---------------- B-matrix prepack into WMMA lane order ----------------------
// For k-step s and lane l:  n = l&15, half = l>>4, k0 = 4*s' + 2*half where
// s' is the step within the weight's own K-range. Stored as float2 per lane so
// transforms do one coalesced b64 load per step, already zero-padded for n>=H1.
__global__ void prep_wpack(const float* __restrict__ W1l, const float* __restrict__ W1r,
                           const float* __restrict__ W2l, const float* __restrict__ W2r,
                           float* __restrict__ Bpack1,    // 8 steps * 32 lanes * 2
                           float* __restrict__ Bpack2) {  // 4 steps * 32 lanes * 2
    int t = threadIdx.x;            // single block of 256
    {
        int step = t >> 5;          // 0..7
        int lane = t & 31;
        int n = lane & 15, half = lane >> 4;
        int sp = step & 3;          // step within the 16-wide K chunk
        int k0 = 4 * sp + 2 * half;
        const float* W = (step < 4) ? W1l : W1r;
        float bx = (n < H1) ? W[n * F_IN + k0]     : 0.0f;
        float by = (n < H1) ? W[n * F_IN + k0 + 1] : 0.0f;
        Bpack1[t * 2 + 0] = bx;
        Bpack1[t * 2 + 1] = by;
    }
    if (t < 128) {
        int step = t >> 5;          // 0..3
        int lane = t & 31;
        int n = lane & 15, half = lane >> 4;
        int sp = step & 1;          // step within the 8-wide K chunk
        int k0 = 4 * sp + 2 * half;
        const float* W = (step < 2) ? W2l : W2r;
        Bpack2[t * 2 + 0] = W[n * H1 + k0];
        Bpack2[t * 2 + 1] = W[n * H1 + k0 + 1];
    }
}

// ---------------- edge scatter (mean aggregation numerators) -----------------
__global__ void edge_agg1(const float* __restrict__ x, const int* __restrict__ ei,
                          float* __restrict__ agg1, float* __restrict__ deg) {
    int e = blockIdx.x * blockDim.x + threadIdx.x;
    if (e >= N_EDGES) return;
    int s = ei[e];
    int d = ei[N_EDGES + e];
    const float4* xr = reinterpret_cast<const float4*>(x + (size_t)s * F_IN);
    float v[16];
    *reinterpret_cast<float4*>(v + 0)  = xr[0];
    *reinterpret_cast<float4*>(v + 4)  = xr[1];
    *reinterpret_cast<float4*>(v + 8)  = xr[2];
    *reinterpret_cast<float4*>(v + 12) = xr[3];
    float* ar = agg1 + (size_t)d * F_IN;
#pragma unroll
    for (int j = 0; j < 16; ++j) atomicAdd(ar + j, v[j]);
    atomicAdd(deg + d, 1.0f);
}

__global__ void edge_agg2(const float* __restrict__ h1, const int* __restrict__ ei,
                          float* __restrict__ agg2) {
    int e = blockIdx.x * blockDim.x + threadIdx.x;
    if (e >= N_EDGES) return;
    int s = ei[e];
    int d = ei[N_EDGES + e];
    const float4* hr = reinterpret_cast<const float4*>(h1 + (size_t)s * H1);
    float v[8];
    *reinterpret_cast<float4*>(v + 0) = hr[0];
    *reinterpret_cast<float4*>(v + 4) = hr[1];
    float* ar = agg2 + (size_t)d * H1;
#pragma unroll
    for (int j = 0; j < 8; ++j) atomicAdd(ar + j, v[j]);
}

// ---------------- SAGEConv node transforms via WMMA f32 16x16x4 --------------
// A-matrix layout (ISA 7.12.2, 32-bit 16x4): lanes 0-15 -> (M=lane, K=k0,k0+1),
// lanes 16-31 -> (M=lane-16, K=k0+2,k0+3). C/D: VGPR r -> M=r (lo half) /
// M=r+8 (hi half), N = lane&15.

__global__ void sage_xform1(const float* __restrict__ x, const float* __restrict__ agg1,
                            const float* __restrict__ deg, const float* __restrict__ b1l,
                            const float* __restrict__ Bpack1, float* __restrict__ h1) {
    int wave = blockIdx.x * (blockDim.x >> 5) + (threadIdx.x >> 5);
    if (wave >= N_TILES) return;           // wave-uniform: EXEC stays all-ones
    int lane = threadIdx.x & 31;
    int row  = lane & 15;
    int half = lane >> 4;
    int m    = wave * 16 + row;            // node index this lane feeds into A
    int n    = row;                        // output column this lane feeds into B
    float inv = 1.0f / fmaxf(deg[m], 1.0f);
    const v2f* __restrict__ bp = (const v2f*)Bpack1;

    v8f acc = {};
    // K = 0..15: mean-aggregated neighbor features through W1l
#pragma unroll
    for (int s = 0; s < 4; ++s) {
        int k0 = 4 * s + 2 * half;
        v2f a;
        a.x = agg1[m * F_IN + k0]     * inv;
        a.y = agg1[m * F_IN + k0 + 1] * inv;
        v2f b = bp[s * 32 + lane];
        acc = __builtin_amdgcn_wmma_f32_16x16x4_f32(false, a, false, b,
                                                    (short)0, acc, false, false);
    }
    // K = 16..31: root features through W1r
#pragma unroll
    for (int s = 0; s < 4; ++s) {
        int k0 = 4 * s + 2 * half;
        v2f a;
        a.x = x[m * F_IN + k0];
        a.y = x[m * F_IN + k0 + 1];
        v2f b = bp[(4 + s) * 32 + lane];
        acc = __builtin_amdgcn_wmma_f32_16x16x4_f32(false, a, false, b,
                                                    (short)0, acc, false, false);
    }
    // epilogue: bias + ReLU, keep only N < H1
    if (n < H1) {
        float bias = b1l[n];
#pragma unroll
        for (int r = 0; r < 8; ++r) {
            int mm = wave * 16 + r + 8 * half;
            h1[mm * H1 + n] = fmaxf(acc[r] + bias, 0.0f);
        }
    }
}

__global__ void sage_xform2(const float* __restrict__ h1, const float* __restrict__ agg2,
                            const float* __restrict__ deg, const float* __restrict__ b2l,
                            const float* __restrict__ Bpack2, float* __restrict__ h2) {
    int wave = blockIdx.x * (blockDim.x >> 5) + (threadIdx.x >> 5);
    if (wave >= N_TILES) return;
    int lane = threadIdx.x & 31;
    int row  = lane & 15;
    int half = lane >> 4;
    int m    = wave * 16 + row;
    int n    = row;
    float inv = 1.0f / fmaxf(deg[m], 1.0f);
    const v2f* __restrict__ bp = (const v2f*)Bpack2;

    v8f acc = {};
    // K = 0..7: mean-aggregated h1 through W2l
#pragma unroll
    for (int s = 0; s < 2; ++s) {
        int k0 = 4 * s + 2 * half;
        v2f a;
        a.x = agg2[m * H1 + k0]     * inv;
        a.y = agg2[m * H1 + k0 + 1] * inv;
        v2f b = bp[s * 32 + lane];
        acc = __builtin_amdgcn_wmma_f32_16x16x4_f32(false, a, false, b,
                                                    (short)0, acc, false, false);
    }
    // K = 8..15: root h1 through W2r
#pragma unroll
    for (int s = 0; s < 2; ++s) {
        int k0 = 4 * s + 2 * half;
        v2f a;
        a.x = h1[m * H1 + k0];
        a.y = h1[m * H1 + k0 + 1];
        v2f b = bp[(2 + s) * 32 + lane];
        acc = __builtin_amdgcn_wmma_f32_16x16x4_f32(false, a, false, b,
                                                    (short)0, acc, false, false);
    }
    float bias = b2l[n];
#pragma unroll
    for (int r = 0; r < 8; ++r) {
        int mm = wave * 16 + r + 8 * half;
        h2[mm * H2 + n] = fmaxf(acc[r] + bias, 0.0f);
    }
}

// ---------------- fused Set2Set + final FC (one block per graph) -------------
__device__ __forceinline__ float sigf(float v) { return 1.0f / (1.0f + __expf(-v)); }

__global__ void set2set_fc(const float* __restrict__ h2,
                           const int* __restrict__ gstart, const int* __restrict__ gend,
                           const float* __restrict__ Wih, const float* __restrict__ Whh,
                           const float* __restrict__ bih, const float* __restrict__ bhh,
                           const float* __restrict__ Wfc, const float* __restrict__ bfc,
                           float* __restrict__ out) {
    __shared__ float sh[H2], sc[H2], sq[2 * H2], sg[4 * H2];
    __shared__ float red[256];
    __shared__ float sacc[H2 + 1];   // [0..15] = r numerator, [16] = softmax denom

    const int g   = blockIdx.x;
    const int tid = threadIdx.x;
    if (tid < H2)     { sh[tid] = 0.0f; sc[tid] = 0.0f; }
    if (tid < 2 * H2) sq[tid] = 0.0f;
    __syncthreads();

    const int s0 = gstart[g];
    const int s1 = gend[g];      // empty graph: s0 = N_NODES, s1 = 0 -> loops skip

    for (int step = 0; step < STEPS; ++step) {
        // LSTM gates: gates = q_star @ Wih^T + bih + h @ Whh^T + bhh
        if (tid < 4 * H2) {
            float a = bih[tid] + bhh[tid];
#pragma unroll
            for (int k = 0; k < 2 * H2; ++k) a += sq[k] * Wih[tid * (2 * H2) + k];
#pragma unroll
            for (int k = 0; k < H2; ++k)     a += sh[k] * Whh[tid * H2 + k];
            sg[tid] = a;
        }
        __syncthreads();
        if (tid < H2) {  // torch gate order (i, f, g, o)
            float ig = sigf(sg[tid]);
            float fg = sigf(sg[H2 + tid]);
            float gg = tanhf(sg[2 * H2 + tid]);
            float og = sigf(sg[3 * H2 + tid]);
            float cn = fg * sc[tid] + ig * gg;
            sc[tid] = cn;
            sh[tid] = og * tanhf(cn);
        }
        __syncthreads();

        // pass 1: segment max of e_n = <h2[n], q>
        float lmax = -3.402823e38f;
        for (int nd = s0 + tid; nd < s1; nd += 256) {
            const float* xr = h2 + (size_t)nd * H2;
            float e = 0.0f;
#pragma unroll
            for (int k = 0; k < H2; ++k) e += xr[k] * sh[k];
            lmax = fmaxf(lmax, e);
        }
        red[tid] = lmax;
        __syncthreads();
        for (int s = 128; s > 0; s >>= 1) {
            if (tid < s) red[tid] = fmaxf(red[tid], red[tid + s]);
            __syncthreads();
        }
        const float emax = red[0];

        if (tid < H2 + 1) sacc[tid] = 0.0f;
        __syncthreads();

        // pass 2: softmax-weighted readout r
        float lr[H2];
#pragma unroll
        for (int k = 0; k < H2; ++k) lr[k] = 0.0f;
        float lsum = 0.0f;
        for (int nd = s0 + tid; nd < s1; nd += 256) {
            const float* xr = h2 + (size_t)nd * H2;
            float e = 0.0f;
#pragma unroll
            for (int k = 0; k < H2; ++k) e += xr[k] * sh[k];
            float a = __expf(e - emax);
            lsum += a;
#pragma unroll
            for (int k = 0; k < H2; ++k) lr[k] += a * xr[k];
        }
        if (s0 + tid < s1) {   // only contributing threads hit shared atomics
            atomicAdd(&sacc[H2], lsum);
#pragma unroll
            for (int k = 0; k < H2; ++k) atomicAdd(&sacc[k], lr[k]);
        }
        __syncthreads();

        if (tid < H2) {
            float asum = sacc[H2];
            sq[tid]      = sh[tid];
            sq[H2 + tid] = (asum > 0.0f) ? sacc[tid] / asum : 0.0f;
        }
        __syncthreads();
    }

    // final fc: out = q_star @ Wfc^T + bfc   (2 outputs per graph)
    if (tid < 2) {
        float a = bfc[tid];
#pragma unroll
        for (int k = 0; k < 2 * H2; ++k) a += sq[k] * Wfc[tid * (2 * H2) + k];
        out[g * 2 + tid] = a;
    }
}

// ---------------------------------------------------------------------------
extern "C" void kernel_launch(void* const* d_in, const int* in_sizes, int n_in,
                              void* d_out, int out_size, void* d_ws, size_t ws_size,
                              hipStream_t stream) {
    const float* x    = (const float*)d_in[0];
    const int*   ei   = (const int*)d_in[1];     // [2, E] flat: src then dst
    const int*   bat  = (const int*)d_in[2];
    const float* W1l  = (const float*)d_in[3];
    const float* b1l  = (const float*)d_in[4];
    const float* W1r  = (const float*)d_in[5];
    const float* W2l  = (const float*)d_in[6];
    const float* b2l  = (const float*)d_in[7];
    const float* W2r  = (const float*)d_in[8];
    const float* Wih  = (const float*)d_in[9];
    const float* Whh  = (const float*)d_in[10];
    const float* bih  = (const float*)d_in[11];
    const float* bhh  = (const float*)d_in[12];
    const float* Wfc  = (const float*)d_in[13];
    const float* bfc  = (const float*)d_in[14];
    float* out = (float*)d_out;

    // workspace layout (floats): [agg1 | deg | agg2] (zeroed) | h1 | h2 | ints | Bpacks
    float* agg1 = (float*)d_ws;                    // N*16
    float* deg  = agg1 + (size_t)N_NODES * F_IN;   // N
    float* agg2 = deg  + N_NODES;                  // N*8
    float* h1   = agg2 + (size_t)N_NODES * H1;     // N*8
    float* h2   = h1   + (size_t)N_NODES * H1;     // N*16
    int* gstart = (int*)(h2 + (size_t)N_NODES * H2);
    int* gend   = gstart + N_GRAPHS;
    float* Bpack1 = (float*)(gend + N_GRAPHS);     // 512 floats (8B-aligned)
    float* Bpack2 = Bpack1 + 8 * 32 * 2;           // 256 floats

    const int nz = N_NODES * F_IN + N_NODES + N_NODES * H1;  // 5,000,000
    zero_f<<<(nz + 255) / 256, 256, 0, stream>>>(agg1, nz);
    init_seg<<<(N_GRAPHS + 255) / 256, 256, 0, stream>>>(gstart, gend);
    seg_bounds<<<(N_NODES + 255) / 256, 256, 0, stream>>>(bat, gstart, gend);
    prep_wpack<<<1, 256, 0, stream>>>(W1l, W1r, W2l, W2r, Bpack1, Bpack2);

    edge_agg1<<<(N_EDGES + 255) / 256, 256, 0, stream>>>(x, ei, agg1, deg);

    const int xblocks = (N_TILES + 7) / 8;   // 8 waves (tiles) per 256-thread block
    sage_xform1<<<xblocks, 256, 0, stream>>>(x, agg1, deg, b1l, Bpack1, h1);

    edge_agg2<<<(N_EDGES + 255) / 256, 256, 0, stream>>>(h1, ei, agg2);

    sage_xform2<<<xblocks, 256, 0, stream>>>(h1, agg2, deg, b2l, Bpack2, h2);

    set2set_fc<<<N_GRAPHS, 256, 0, stream>>>(h2, gstart, gend, Wih, Whh, bih, bhh,
                                             Wfc, bfc, out);
}